// GNNModel_30477087932519
// MI455X (gfx1250) — compile-verified
//
#include <hip/hip_runtime.h>
#include <hip/hip_bf16.h>
#include <math.h>

typedef __attribute__((ext_vector_type(2))) float v2f;
typedef __attribute__((ext_vector_type(8))) float v8f;

#define NNODES 200000
#define NEDGES 600000
#define NGRAPH 512
#define BN_EPS 1e-5f

// ---- workspace layout (float offsets, all 64-float aligned) ----
#define OFF_DIS    ((size_t)0)          // N        (deg -> dis in place)
#define OFF_H1     ((size_t)200704)     // N*64
#define OFF_AGG1   ((size_t)13000704)   // N*64
#define OFF_H2     ((size_t)25800704)   // N*128
#define OFF_AGG2   ((size_t)51400704)   // N*128
#define OFF_POOLED ((size_t)77000704)   // G*128
#define OFF_T1     ((size_t)77066240)   // G*256
#define OFF_XF     ((size_t)77197312)   // G*128
#define OFF_C1     ((size_t)77262848)   // G*192
#define OFF_C2     ((size_t)77361152)   // G*128

// -------------------------------------------------------------------------
__global__ void zero_f32_kernel(float* __restrict__ p, int n) {
    int i = blockIdx.x * blockDim.x + threadIdx.x;
    if (i < n) p[i] = 0.0f;
}

// deg[dst] += 1 per edge
__global__ void deg_kernel(const int* __restrict__ ei, float* __restrict__ deg) {
    int e = blockIdx.x * blockDim.x + threadIdx.x;
    if (e >= NEDGES) return;
    atomicAdd(&deg[ei[NEDGES + e]], 1.0f);
}

// dis = (deg + 1)^-0.5, in place
__global__ void dis_kernel(float* __restrict__ dis) {
    int n = blockIdx.x * blockDim.x + threadIdx.x;
    if (n >= NNODES) return;
    dis[n] = rsqrtf(dis[n] + 1.0f);
}

// h1 = x @ W1 (K=2); agg1 initialized with self-loop term + bias
__global__ void gcn1_lin_kernel(const float* __restrict__ x,
                                const float* __restrict__ W1,
                                const float* __restrict__ b1,
                                const float* __restrict__ dis,
                                float* __restrict__ h1,
                                float* __restrict__ agg1) {
    int idx = blockIdx.x * blockDim.x + threadIdx.x;
    if (idx >= NNODES * 64) return;
    int n = idx >> 6, j = idx & 63;
    float x0 = x[2 * n], x1 = x[2 * n + 1];
    float v = x0 * W1[j] + x1 * W1[64 + j];
    h1[idx] = v;
    float d = dis[n];
    agg1[idx] = v * d * d + b1[j];
}

// layer-1 scatter: one wave per edge, 64 feats = 32 lanes x float2
__global__ void gcn1_edge_kernel(const int* __restrict__ ei,
                                 const float* __restrict__ dis,
                                 const float* __restrict__ h1,
                                 float* __restrict__ agg1) {
    int e = blockIdx.x * 8 + (threadIdx.x >> 5);
    int lane = threadIdx.x & 31;
    int s = ei[e], d = ei[NEDGES + e];
    float w = dis[s] * dis[d];
    const float2 m = *(const float2*)(h1 + (size_t)s * 64 + lane * 2);
    float* out = agg1 + (size_t)d * 64 + lane * 2;
    atomicAdd(out + 0, m.x * w);
    atomicAdd(out + 1, m.y * w);
}

// GEMM: h2 = relu(agg1) @ W2  ([200000,64]x[64,128]) via V_WMMA_F32_16X16X4_F32.
// W2 staged in LDS (32 KB); B fragments batch-preloaded per n-tile so the 16
// WMMAs issue back-to-back behind a single wait. ReLU fused into the A load.
// Epilogue also initializes agg2 = h2*dis^2 + b2 (self-loop + bias).
__global__ void __launch_bounds__(256)
gcn2_gemm_kernel(const float* __restrict__ a1,
                 const float* __restrict__ W2,
                 const float* __restrict__ b2,
                 const float* __restrict__ dis,
                 float* __restrict__ h2,
                 float* __restrict__ agg2) {
    __shared__ float sW2[64 * 128];
    {   // cooperative W2 stage: 8192 floats as 2048 float4s, 256 threads
        const float4* src = (const float4*)W2;
        float4* dst = (float4*)sW2;
#pragma unroll
        for (int i = 0; i < 8; ++i)
            dst[threadIdx.x + i * 256] = src[threadIdx.x + i * 256];
    }
    __syncthreads();

    const int wave = threadIdx.x >> 5;
    const int lane = threadIdx.x & 31;
    const int tile = blockIdx.x * 8 + wave;
    if (tile >= NNODES / 16) return;
    const int r0   = tile * 16;
    const int half = lane >> 4;   // 0: K pair {0,1}; 1: K pair {2,3}
    const int l    = lane & 15;

    // preload all 16 A fragments (K=64 in slices of 4) as b64, ReLU fused
    v2f af[16];
    const v2f* ap = (const v2f*)(a1 + (size_t)(r0 + l) * 64 + half * 2);
#pragma unroll
    for (int k = 0; k < 16; ++k) {
        v2f a = ap[k * 2];
        af[k].x = fmaxf(a.x, 0.0f);
        af[k].y = fmaxf(a.y, 0.0f);
    }

    // hoist self-loop scale factors for this tile's 8 rows per half
    float dsc[8];
#pragma unroll
    for (int r = 0; r < 8; ++r) {
        float d = dis[r0 + half * 8 + r];
        dsc[r] = d * d;
    }

    const float* bbase = sW2 + (half * 2) * 128 + l;
#pragma unroll
    for (int nt = 0; nt < 8; ++nt) {
        const int c0 = nt * 16;
        // batch-load the 16 B fragments for this n-tile from LDS
        v2f bf[16];
#pragma unroll
        for (int k = 0; k < 16; ++k) {
            const float* bp = bbase + c0 + k * 4 * 128;
            bf[k].x = bp[0];
            bf[k].y = bp[128];
        }
        v8f acc = {};
#pragma unroll
        for (int k = 0; k < 16; ++k)
            acc = __builtin_amdgcn_wmma_f32_16x16x4_f32(
                false, af[k], false, bf[k], (short)0, acc, false, false);

        const float bias = b2[c0 + l];
#pragma unroll
        for (int r = 0; r < 8; ++r) {
            const int row = r0 + half * 8 + r;   // C/D layout: lanes>=16 hold M=8..15
            const size_t idx = (size_t)row * 128 + c0 + l;
            const float v = acc[r];
            h2[idx] = v;
            agg2[idx] = v * dsc[r] + bias;
        }
    }
}

// layer-2 scatter: one wave per edge, 128 feats = 32 lanes x float4
__global__ void gcn2_edge_kernel(const int* __restrict__ ei,
                                 const float* __restrict__ dis,
                                 const float* __restrict__ h2,
                                 float* __restrict__ agg2) {
    int e = blockIdx.x * 8 + (threadIdx.x >> 5);
    int lane = threadIdx.x & 31;
    int s = ei[e], d = ei[NEDGES + e];
    float w = dis[s] * dis[d];
    const float4 m = *(const float4*)(h2 + (size_t)s * 128 + lane * 4);
    float* out = agg2 + (size_t)d * 128 + lane * 4;
    atomicAdd(out + 0, m.x * w);
    atomicAdd(out + 1, m.y * w);
    atomicAdd(out + 2, m.z * w);
    atomicAdd(out + 3, m.w * w);
}

__device__ inline int lower_bound_i(const int* a, int n, int key) {
    int lo = 0, hi = n;
    while (lo < hi) {
        int mid = (lo + hi) >> 1;
        if (a[mid] < key) lo = mid + 1; else hi = mid;
    }
    return lo;
}

// global mean pool: batch is SORTED -> binary-search each graph's node range.
// One 128-thread block per graph; thread t owns feature t. ReLU applied here.
__global__ void pool_kernel(const float* __restrict__ agg2,
                            const int* __restrict__ batch,
                            float* __restrict__ pooled) {
    const int g = blockIdx.x;
    const int f = threadIdx.x;                 // 0..127
    const int lo = lower_bound_i(batch, NNODES, g);
    const int hi = lower_bound_i(batch, NNODES, g + 1);
    float s = 0.0f;
    for (int n = lo; n < hi; ++n)
        s += fmaxf(agg2[(size_t)n * 128 + f], 0.0f);
    const float cnt = (float)(hi - lo);
    pooled[g * 128 + f] = s / fmaxf(cnt, 1.0f);
}

// feat MLP stage 1: t1 = relu(BN(feat @ fW1 + fb1))   [G,256]
__global__ void featmlp1_kernel(const float* __restrict__ feat,
                                const float* __restrict__ fW1,
                                const float* __restrict__ fb1,
                                const float* __restrict__ g1,
                                const float* __restrict__ be1,
                                const float* __restrict__ rm1,
                                const float* __restrict__ rv1,
                                float* __restrict__ t1) {
    int idx = blockIdx.x * blockDim.x + threadIdx.x;
    if (idx >= NGRAPH * 256) return;
    int gi = idx >> 8, j = idx & 255;
    const float* f = feat + gi * 8;
    float s = fb1[j];
#pragma unroll
    for (int k = 0; k < 8; ++k) s += f[k] * fW1[k * 256 + j];
    s = (s - rm1[j]) * rsqrtf(rv1[j] + BN_EPS) * g1[j] + be1[j];
    t1[idx] = fmaxf(s, 0.0f);
}

// feat MLP stage 2: xf = t1 @ fW2 + fb2   [G,128]
__global__ void featmlp2_kernel(const float* __restrict__ t1,
                                const float* __restrict__ fW2,
                                const float* __restrict__ fb2,
                                float* __restrict__ xf) {
    int idx = blockIdx.x * blockDim.x + threadIdx.x;
    if (idx >= NGRAPH * 128) return;
    int gi = idx >> 7, j = idx & 127;
    const float* t = t1 + gi * 256;
    float s = fb2[j];
    for (int k = 0; k < 256; ++k) s += t[k] * fW2[k * 128 + j];
    xf[idx] = s;
}

// fusion stage 1: c1 = relu(BN([pooled|xf] @ uW1 + ub1))   [G,192]
__global__ void fuse1_kernel(const float* __restrict__ pooled,
                             const float* __restrict__ xf,
                             const float* __restrict__ uW1,
                             const float* __restrict__ ub1,
                             const float* __restrict__ g1,
                             const float* __restrict__ be1,
                             const float* __restrict__ rm1,
                             const float* __restrict__ rv1,
                             float* __restrict__ c1) {
    int idx = blockIdx.x * blockDim.x + threadIdx.x;
    if (idx >= NGRAPH * 192) return;
    int gi = idx / 192, j = idx - gi * 192;
    const float* p = pooled + gi * 128;
    const float* q = xf + gi * 128;
    float s = ub1[j];
    for (int k = 0; k < 128; ++k) s += p[k] * uW1[k * 192 + j];
    for (int k = 0; k < 128; ++k) s += q[k] * uW1[(128 + k) * 192 + j];
    s = (s - rm1[j]) * rsqrtf(rv1[j] + BN_EPS) * g1[j] + be1[j];
    c1[idx] = fmaxf(s, 0.0f);
}

// fusion stage 2: c2 = relu(BN(c1 @ uW2 + ub2))   [G,128]
__global__ void fuse2_kernel(const float* __restrict__ c1,
                             const float* __restrict__ uW2,
                             const float* __restrict__ ub2,
                             const float* __restrict__ g2,
                             const float* __restrict__ be2,
                             const float* __restrict__ rm2,
                             const float* __restrict__ rv2,
                             float* __restrict__ c2) {
    int idx = blockIdx.x * blockDim.x + threadIdx.x;
    if (idx >= NGRAPH * 128) return;
    int gi = idx >> 7, j = idx & 127;
    const float* c = c1 + gi * 192;
    float s = ub2[j];
    for (int k = 0; k < 192; ++k) s += c[k] * uW2[k * 128 + j];
    s = (s - rm2[j]) * rsqrtf(rv2[j] + BN_EPS) * g2[j] + be2[j];
    c2[idx] = fmaxf(s, 0.0f);
}

// fusion stage 3: out = sigmoid(c2 @ uW3 + ub3)   [G]
__global__ void fuse3_kernel(const float* __restrict__ c2,
                             const float* __restrict__ uW3,
                             const float* __restrict__ ub3,
                             float* __restrict__ out) {
    int g = blockIdx.x * blockDim.x + threadIdx.x;
    if (g >= NGRAPH) return;
    const float* c = c2 + g * 128;
    float s = ub3[0];
    for (int k = 0; k < 128; ++k) s += c[k] * uW3[k];
    out[g] = 1.0f / (1.0f + expf(-s));
}

// -------------------------------------------------------------------------
extern "C" void kernel_launch(void* const* d_in, const int* in_sizes, int n_in,
                              void* d_out, int out_size, void* d_ws, size_t ws_size,
                              hipStream_t stream) {
    const float* x    = (const float*)d_in[0];
    const float* feat = (const float*)d_in[1];
    const float* W1   = (const float*)d_in[2];
    const float* b1   = (const float*)d_in[3];
    const float* W2   = (const float*)d_in[4];
    const float* b2   = (const float*)d_in[5];
    const float* fW1  = (const float*)d_in[6];
    const float* fb1  = (const float*)d_in[7];
    const float* fg1  = (const float*)d_in[8];
    const float* fbe1 = (const float*)d_in[9];
    const float* frm1 = (const float*)d_in[10];
    const float* frv1 = (const float*)d_in[11];
    const float* fW2  = (const float*)d_in[12];
    const float* fb2  = (const float*)d_in[13];
    const float* uW1  = (const float*)d_in[14];
    const float* ub1  = (const float*)d_in[15];
    const float* ug1  = (const float*)d_in[16];
    const float* ube1 = (const float*)d_in[17];
    const float* urm1 = (const float*)d_in[18];
    const float* urv1 = (const float*)d_in[19];
    const float* uW2  = (const float*)d_in[20];
    const float* ub2  = (const float*)d_in[21];
    const float* ug2  = (const float*)d_in[22];
    const float* ube2 = (const float*)d_in[23];
    const float* urm2 = (const float*)d_in[24];
    const float* urv2 = (const float*)d_in[25];
    const float* uW3  = (const float*)d_in[26];
    const float* ub3  = (const float*)d_in[27];
    const int* ei     = (const int*)d_in[28];
    const int* batch  = (const int*)d_in[29];
    float* out = (float*)d_out;

    float* ws     = (float*)d_ws;
    float* dis    = ws + OFF_DIS;
    float* h1     = ws + OFF_H1;
    float* agg1   = ws + OFF_AGG1;
    float* h2     = ws + OFF_H2;
    float* agg2   = ws + OFF_AGG2;
    float* pooled = ws + OFF_POOLED;
    float* t1     = ws + OFF_T1;
    float* xf     = ws + OFF_XF;
    float* c1     = ws + OFF_C1;
    float* c2     = ws + OFF_C2;

    // degrees -> symmetric norm
    zero_f32_kernel<<<(NNODES + 255) / 256, 256, 0, stream>>>(dis, NNODES);
    deg_kernel<<<(NEDGES + 255) / 256, 256, 0, stream>>>(ei, dis);
    dis_kernel<<<(NNODES + 255) / 256, 256, 0, stream>>>(dis);

    // GCN layer 1 (2 -> 64)
    gcn1_lin_kernel<<<(NNODES * 64) / 256, 256, 0, stream>>>(x, W1, b1, dis, h1, agg1);
    gcn1_edge_kernel<<<NEDGES / 8, 256, 0, stream>>>(ei, dis, h1, agg1);

    // GCN layer 2 (64 -> 128): WMMA GEMM (ReLU of layer-1 fused into A load)
    gcn2_gemm_kernel<<<(NNODES / 16 + 7) / 8, 256, 0, stream>>>(agg1, W2, b2, dis, h2, agg2);
    gcn2_edge_kernel<<<NEDGES / 8, 256, 0, stream>>>(ei, dis, h2, agg2);

    // mean pool (sorted batch, atomic-free)
    pool_kernel<<<NGRAPH, 128, 0, stream>>>(agg2, batch, pooled);

    // feat MLP
    featmlp1_kernel<<<(NGRAPH * 256) / 256, 256, 0, stream>>>(
        feat, fW1, fb1, fg1, fbe1, frm1, frv1, t1);
    featmlp2_kernel<<<(NGRAPH * 128) / 256, 256, 0, stream>>>(t1, fW2, fb2, xf);

    // fusion head
    fuse1_kernel<<<(NGRAPH * 192) / 256, 256, 0, stream>>>(
        pooled, xf, uW1, ub1, ug1, ube1, urm1, urv1, c1);
    fuse2_kernel<<<(NGRAPH * 128) / 256, 256, 0, stream>>>(
        c1, uW2, ub2, ug2, ube2, urm2, urv2, c2);
    fuse3_kernel<<<(NGRAPH + 255) / 256, 256, 0, stream>>>(c2, uW3, ub3, out);
}